// Diversity_53833120088162
// MI455X (gfx1250) — compile-verified
//
#include <hip/hip_runtime.h>

typedef __attribute__((ext_vector_type(2))) float v2f;
typedef __attribute__((ext_vector_type(8))) float v8f;

#define N_ROWS 16384
#define M_ROWS 8192
#define DDIM   256
#define TCHAINS 4                           // 16-row tiles (indep. WMMA chains) per wave
#define ROWS_PER_WG (TCHAINS * 16)          // 64
#define LDS_STRIDE  260   // 256 + 4 pad floats -> conflict-free ds_load_b64 pattern
#define TILES_PER_WAVE (M_ROWS / 16 / 8)    // 64 column tiles per wave

// ---------------------------------------------------------------------------
// Kernel 1: row squared-norms for U (first N rows) and L (next M rows).
// ---------------------------------------------------------------------------
__global__ void norms_kernel(const float* __restrict__ U,
                             const float* __restrict__ L,
                             float* __restrict__ u2,
                             float* __restrict__ l2) {
    int idx = blockIdx.x * blockDim.x + threadIdx.x;   // 0 .. N+M-1
    const float* src;
    float* dst;
    if (idx < N_ROWS) {
        src = U + (size_t)idx * DDIM;
        dst = u2 + idx;
    } else {
        int r = idx - N_ROWS;
        src = L + (size_t)r * DDIM;
        dst = l2 + r;
    }
    float s = 0.0f;
    const float4* s4 = (const float4*)src;
#pragma unroll
    for (int i = 0; i < DDIM / 4; ++i) {
        float4 v = s4[i];
        s = fmaf(v.x, v.x, s);
        s = fmaf(v.y, v.y, s);
        s = fmaf(v.z, v.z, s);
        s = fmaf(v.w, v.w, s);
    }
    *dst = s;
}

// ---------------------------------------------------------------------------
// Kernel 2: fused distance GEMM + row-min + sqrt.
// Grid: N/64 workgroups, 256 threads (8 waves).
// All 8 waves cover the SAME 64 rows (four 16-row tiles -> four independent
// accumulator chains) and split the M column tiles into eighths. Each B
// fragment is loaded once per WG and feeds FOUR v_wmma chains (4x ILP,
// 1/4 the B traffic of the naive scheme).
// ---------------------------------------------------------------------------
__global__ void __launch_bounds__(256)
dist_min_kernel(const float* __restrict__ U,
                const float* __restrict__ L,
                const float* __restrict__ u2,
                const float* __restrict__ l2,
                float* __restrict__ out) {
    extern __shared__ float smem[];
    float* sU  = smem;                                  // 64 x 260 floats
    float* red = smem + ROWS_PER_WG * LDS_STRIDE;       // 7 x 64 partial mins

    const int tid  = threadIdx.x;
    const int lane = tid & 31;
    const int wave = tid >> 5;
    const int lrow = lane & 15;   // N-index within tile (and row index for A/B frags)
    const int lhi  = lane >> 4;   // lane-half -> K offset 0/2, and M offset 0/8 in C

    const int r0 = blockIdx.x * ROWS_PER_WG;            // global base row

    // ---- stage this WG's 64 U rows into LDS (coalesced float4) ----
    {
        const int row = tid >> 2;          // 0..63
        const int cb  = (tid & 3) * 64;    // column base within row
        const float4* g = (const float4*)(U + (size_t)(r0 + row) * DDIM + cb);
        float4* s = (float4*)(sU + row * LDS_STRIDE + cb);
#pragma unroll
        for (int i = 0; i < 16; ++i) s[i] = g[i];
    }
    __syncthreads();

    // A fragment bases for the four 16-row tiles
    const float* aBase = sU + lrow * LDS_STRIDE + 2 * lhi;

    float vmin[TCHAINS][8];
#pragma unroll
    for (int t = 0; t < TCHAINS; ++t)
#pragma unroll
        for (int p = 0; p < 8; ++p) vmin[t][p] = __builtin_inff();

    const int ct0 = wave * TILES_PER_WAVE;
    const int ct1 = ct0 + TILES_PER_WAVE;
    for (int ct = ct0; ct < ct1; ++ct) {
        // B fragment: L row (ct*16 + lrow), same per-lane K pattern as A
        const float* bBase = L + (size_t)(ct * 16 + lrow) * DDIM + 2 * lhi;
        v8f acc[TCHAINS];
#pragma unroll
        for (int t = 0; t < TCHAINS; ++t)
            acc[t] = (v8f){0.f, 0.f, 0.f, 0.f, 0.f, 0.f, 0.f, 0.f};

#pragma unroll 8
        for (int k = 0; k < DDIM; k += 4) {
            v2f b = *(const v2f*)(bBase + k);   // global_load_b64 (L2-resident), shared x4
#pragma unroll
            for (int t = 0; t < TCHAINS; ++t) {
                v2f a = *(const v2f*)(aBase + t * (16 * LDS_STRIDE) + k); // ds_load_b64
                acc[t] = __builtin_amdgcn_wmma_f32_16x16x4_f32(
                    false, a, false, b, (short)0, acc[t], false, false);
            }
        }
        // min term: l2[n] - 2*dot ; u2 added after the min
        const float l2n = l2[ct * 16 + lrow];
#pragma unroll
        for (int t = 0; t < TCHAINS; ++t)
#pragma unroll
            for (int p = 0; p < 8; ++p)
                vmin[t][p] = fminf(vmin[t][p], fmaf(-2.0f, acc[t][p], l2n));
    }

    // ---- min across the 16 lanes holding different N (xor stays in half) ----
#pragma unroll
    for (int t = 0; t < TCHAINS; ++t) {
#pragma unroll
        for (int p = 0; p < 8; ++p) {
            float v = vmin[t][p];
            v = fminf(v, __shfl_xor(v, 1, 32));
            v = fminf(v, __shfl_xor(v, 2, 32));
            v = fminf(v, __shfl_xor(v, 4, 32));
            v = fminf(v, __shfl_xor(v, 8, 32));
            vmin[t][p] = v;   // lanes 0/16 hold mins for M = p / p+8 of tile t
        }
    }

    // ---- combine the 8 per-wave partials via LDS ----
    if (wave > 0 && lrow == 0) {
#pragma unroll
        for (int t = 0; t < TCHAINS; ++t)
#pragma unroll
            for (int p = 0; p < 8; ++p)
                red[(wave - 1) * ROWS_PER_WG + (t * 16 + lhi * 8 + p)] = vmin[t][p];
    }
    __syncthreads();
    if (wave == 0 && lrow == 0) {
#pragma unroll
        for (int t = 0; t < TCHAINS; ++t) {
#pragma unroll
            for (int p = 0; p < 8; ++p) {
                const int li  = t * 16 + lhi * 8 + p;
                float m = vmin[t][p];
#pragma unroll
                for (int j = 0; j < 7; ++j) m = fminf(m, red[j * ROWS_PER_WG + li]);
                const int row = r0 + li;
                out[row] = sqrtf(fmaxf(u2[row] + m, 0.0f));
            }
        }
    }
}

// ---------------------------------------------------------------------------
extern "C" void kernel_launch(void* const* d_in, const int* in_sizes, int n_in,
                              void* d_out, int out_size, void* d_ws, size_t ws_size,
                              hipStream_t stream) {
    (void)in_sizes; (void)n_in; (void)out_size; (void)ws_size;
    // setup_inputs order: pred (unused), U_z, L_z
    const float* U = (const float*)d_in[1];
    const float* L = (const float*)d_in[2];
    float* out = (float*)d_out;

    float* u2 = (float*)d_ws;          // N floats
    float* l2 = u2 + N_ROWS;           // M floats  (total 96 KB of workspace)

    norms_kernel<<<(N_ROWS + M_ROWS) / 256, 256, 0, stream>>>(U, L, u2, l2);

    const size_t shmem = (ROWS_PER_WG * LDS_STRIDE + 7 * ROWS_PER_WG) * sizeof(float);
    dist_min_kernel<<<N_ROWS / ROWS_PER_WG, 256, shmem, stream>>>(U, L, u2, l2, out);
}